// Net_82669530514071
// MI455X (gfx1250) — compile-verified
//
#include <hip/hip_runtime.h>

typedef __attribute__((ext_vector_type(2))) float v2f;
typedef __attribute__((ext_vector_type(8))) float v8f;

#define DEVINL __device__ __forceinline__

namespace {

constexpr int LS = 68;   // padded row stride (floats) for 64-wide matrices: conflict-free b64 frags
constexpr int SS = 12;   // padded row stride for 8-wide matrices
constexpr int O_W12  = 0;
constexpr int O_W13  = O_W12  + 64 * LS;
constexpr int O_W23  = O_W13  + 64 * LS;
constexpr int O_W12T = O_W23  + 64 * LS;
constexpr int O_W13T = O_W12T + 64 * LS;
constexpr int O_W23T = O_W13T + 64 * LS;
constexpr int O_W11  = O_W23T + 64 * LS;
constexpr int O_W22  = O_W11  + 64 * SS;
constexpr int O_W11T = O_W22  + 64 * SS;   // [16][LS], rows 0..7 valid (cols 8..15 of tile unused)
constexpr int O_W22T = O_W11T + 16 * LS;
constexpr int O_B11  = O_W22T + 16 * LS;
constexpr int O_B12  = O_B11 + 64;
constexpr int O_B22  = O_B12 + 64;
constexpr int O_B13  = O_B22 + 64;
constexpr int O_B23  = O_B13 + 64;
constexpr int O_WF   = O_B23 + 64;
constexpr int O_ACT  = O_WF + 64;
constexpr int WAVES  = 8;
constexpr int SMEM_FLOATS = O_ACT + WAVES * 16 * LS;

DEVINL v8f wmma4(v2f a, v2f b, v8f c) {
  // D = A(16x4 f32) * B(4x16 f32) + C(16x16 f32), full fp32 precision
  return __builtin_amdgcn_wmma_f32_16x16x4_f32(false, a, false, b, (short)0, c,
                                               false, false);
}

DEVINL v2f ldv2(const float* p) { return *(const v2f*)p; }

DEVINL void zero4(v8f c[4]) {
#pragma unroll
  for (int t = 0; t < 4; ++t)
#pragma unroll
    for (int j = 0; j < 8; ++j) c[t][j] = 0.0f;
}

DEVINL void initBias(v8f c[4], const float* b, int ln) {
#pragma unroll
  for (int t = 0; t < 4; ++t) {
    float bv = b[t * 16 + ln];
#pragma unroll
    for (int j = 0; j < 8; ++j) c[t][j] = bv;
  }
}

// D-layout store of elementwise scale*a*b into per-wave activation tile [16][LS]
DEVINL void storeProd(float* ab, const v8f a[4], const v8f b[4], int ln, int hi,
                      float scale) {
#pragma unroll
  for (int t = 0; t < 4; ++t)
#pragma unroll
    for (int j = 0; j < 8; ++j)
      ab[(j + 8 * hi) * LS + t * 16 + ln] = scale * a[t][j] * b[t][j];
}

// store a[t][j] * Wf[col] (column-dependent scale) into activation tile
DEVINL void storeScaleCol(float* ab, const v8f a[4], const float* wfs, int ln, int hi) {
#pragma unroll
  for (int t = 0; t < 4; ++t) {
    float w = wfs[t * 16 + ln];
#pragma unroll
    for (int j = 0; j < 8; ++j)
      ab[(j + 8 * hi) * LS + t * 16 + ln] = a[t][j] * w;
  }
}

// acc[N=64] += act(16x64) @ B where B[k][n] = W[n*LS + k]  (16 k-steps)
DEVINL void gemm64(const float* ab, const float* W, int ln, int hi, v8f acc[4]) {
#pragma unroll 4
  for (int kk = 0; kk < 16; ++kk) {
    v2f a = ldv2(ab + ln * LS + kk * 4 + 2 * hi);
#pragma unroll
    for (int t = 0; t < 4; ++t) {
      v2f b = ldv2(W + (t * 16 + ln) * LS + kk * 4 + 2 * hi);
      acc[t] = wmma4(a, b, acc[t]);
    }
  }
}

// two GEMMs sharing the A operand (layer 3 dual branch)
DEVINL void gemm64_dual(const float* ab, const float* Wa, const float* Wb,
                        int ln, int hi, v8f acc1[4], v8f acc2[4]) {
#pragma unroll 2
  for (int kk = 0; kk < 16; ++kk) {
    v2f a = ldv2(ab + ln * LS + kk * 4 + 2 * hi);
#pragma unroll
    for (int t = 0; t < 4; ++t) {
      v2f b1 = ldv2(Wa + (t * 16 + ln) * LS + kk * 4 + 2 * hi);
      acc1[t] = wmma4(a, b1, acc1[t]);
      v2f b2 = ldv2(Wb + (t * 16 + ln) * LS + kk * 4 + 2 * hi);
      acc2[t] = wmma4(a, b2, acc2[t]);
    }
  }
}

// single-output-tile GEMM (N<=16): acc += act(16x64) @ WT (WT row ln, k cols)
DEVINL void gemmN1(const float* ab, const float* WT, int ln, int hi, v8f& acc) {
#pragma unroll 4
  for (int kk = 0; kk < 16; ++kk) {
    v2f a = ldv2(ab + ln * LS + kk * 4 + 2 * hi);
    v2f b = ldv2(WT + ln * LS + kk * 4 + 2 * hi);
    acc = wmma4(a, b, acc);
  }
}

// out[r0+m] = sum_n z13[m][n]*z23[m][n]*Wf[n]
DEVINL void finalDot(const v8f z13[4], const v8f z23[4], const float* wfs,
                     int ln, int hi, float* outp, int r0) {
  float wf[4];
#pragma unroll
  for (int t = 0; t < 4; ++t) wf[t] = wfs[t * 16 + ln];
#pragma unroll
  for (int j = 0; j < 8; ++j) {
    float ssum = 0.0f;
#pragma unroll
    for (int t = 0; t < 4; ++t) ssum += z13[t][j] * z23[t][j] * wf[t];
    ssum += __shfl_xor(ssum, 1, 16);
    ssum += __shfl_xor(ssum, 2, 16);
    ssum += __shfl_xor(ssum, 4, 16);
    ssum += __shfl_xor(ssum, 8, 16);
    if (ln == 0) outp[r0 + j + 8 * hi] = ssum;
  }
}

// out[r0+m] = sum_{d<8} g[m][d] * xdot[r0+m][d]   (g cols 8..15 are garbage -> masked)
DEVINL void gradDot(const v8f& g, const float* xdotg, int r0, int ln, int hi,
                    float* outp) {
#pragma unroll
  for (int j = 0; j < 8; ++j) {
    int m = j + 8 * hi;
    float ssum = (ln < 8) ? g[j] * xdotg[(r0 + m) * 8 + ln] : 0.0f;
    ssum += __shfl_xor(ssum, 1, 16);
    ssum += __shfl_xor(ssum, 2, 16);
    ssum += __shfl_xor(ssum, 4, 16);
    ssum += __shfl_xor(ssum, 8, 16);
    if (ln == 0) outp[r0 + m] = ssum;
  }
}

// Full forward net for one 16-sample tile; z13/z23 are the layer-3 branch
// pre-products. Optionally saves layer-1/2 intermediates for the VJP.
template <bool SAVE>
DEVINL void net_forward(const float* __restrict__ xg, int r0, int ln, int hi,
                        const float* __restrict__ s, float* __restrict__ ab,
                        v8f z13[4], v8f z23[4],
                        v8f zs[4], v8f z1s[4], v8f z2s[4]) {
  // x A-fragments (16x8 tile, two k-steps) straight from global
  v2f ax0 = ldv2(xg + (r0 + ln) * 8 + 2 * hi);
  v2f ax1 = ldv2(xg + (r0 + ln) * 8 + 4 + 2 * hi);

  // layer 1: z = x @ W1_1^T + B1_1
  v8f z[4];
  initBias(z, s + O_B11, ln);
#pragma unroll
  for (int t = 0; t < 4; ++t) {
    v2f b0 = ldv2(s + O_W11 + (t * 16 + ln) * SS + 2 * hi);
    v2f b1 = ldv2(s + O_W11 + (t * 16 + ln) * SS + 4 + 2 * hi);
    z[t] = wmma4(ax0, b0, z[t]);
    z[t] = wmma4(ax1, b1, z[t]);
  }
  if constexpr (SAVE) {
#pragma unroll
    for (int t = 0; t < 4; ++t) zs[t] = z[t];
  }
  storeProd(ab, z, z, ln, hi, 1.0f);  // y = z*z -> LDS

  // layer 2: z1 = y @ W1_2^T + B1_2 ; z2 = x @ W2_2^T + B2_2 ; y2 = z1*z2
  v8f z1[4];
  initBias(z1, s + O_B12, ln);
  gemm64(ab, s + O_W12, ln, hi, z1);
  v8f z2[4];
  initBias(z2, s + O_B22, ln);
#pragma unroll
  for (int t = 0; t < 4; ++t) {
    v2f b0 = ldv2(s + O_W22 + (t * 16 + ln) * SS + 2 * hi);
    v2f b1 = ldv2(s + O_W22 + (t * 16 + ln) * SS + 4 + 2 * hi);
    z2[t] = wmma4(ax0, b0, z2[t]);
    z2[t] = wmma4(ax1, b1, z2[t]);
  }
  if constexpr (SAVE) {
#pragma unroll
    for (int t = 0; t < 4; ++t) { z1s[t] = z1[t]; z2s[t] = z2[t]; }
  }
  storeProd(ab, z1, z2, ln, hi, 1.0f);  // y2 -> LDS

  // layer 3 (dual branch, shared A): z13 = y2 @ W1_3^T + B1_3 ; z23 = y2 @ W2_3^T + B2_3
  initBias(z13, s + O_B13, ln);
  initBias(z23, s + O_B23, ln);
  gemm64_dual(ab, s + O_W13, s + O_W23, ln, hi, z13, z23);
}

__global__ __launch_bounds__(WAVES * 32) void pinn_fused(
    const float* __restrict__ l1g, const float* __restrict__ Ig,
    const float* __restrict__ Ug, const float* __restrict__ ldotg,
    const float* __restrict__ W11g, const float* __restrict__ B11g,
    const float* __restrict__ W12g, const float* __restrict__ B12g,
    const float* __restrict__ W22g, const float* __restrict__ B22g,
    const float* __restrict__ W13g, const float* __restrict__ B13g,
    const float* __restrict__ W23g, const float* __restrict__ B23g,
    const float* __restrict__ Wfg, const float* __restrict__ bm1g,
    float* __restrict__ out, int N) {
  extern __shared__ float s[];
  const int tid = threadIdx.x;

  // ---- stage weights (plus transposed copies for the backward GEMMs) into LDS ----
#pragma unroll 2
  for (int i = tid; i < 64 * 64; i += WAVES * 32) {
    int r = i >> 6, c = i & 63;
    float a = W12g[i]; s[O_W12 + r * LS + c] = a; s[O_W12T + c * LS + r] = a;
    float b = W13g[i]; s[O_W13 + r * LS + c] = b; s[O_W13T + c * LS + r] = b;
    float d = W23g[i]; s[O_W23 + r * LS + c] = d; s[O_W23T + c * LS + r] = d;
  }
  for (int i = tid; i < 64 * 8; i += WAVES * 32) {
    int r = i >> 3, c = i & 7;
    float a = W11g[i]; s[O_W11 + r * SS + c] = a; s[O_W11T + c * LS + r] = a;
    float b = W22g[i]; s[O_W22 + r * SS + c] = b; s[O_W22T + c * LS + r] = b;
  }
  if (tid < 64) {
    s[O_B11 + tid] = B11g[tid];
    s[O_B12 + tid] = B12g[tid];
    s[O_B22 + tid] = B22g[tid];
    s[O_B13 + tid] = B13g[tid];
    s[O_B23 + tid] = B23g[tid];
    s[O_WF + tid] = Wfg[tid];
  }
  __syncthreads();

  const int lane = tid & 31;
  const int wave = tid >> 5;
  const int ln = lane & 15;
  const int hi = lane >> 4;
  const int r0 = (blockIdx.x * WAVES + wave) * 16;
  float* ab = s + O_ACT + wave * 16 * LS;

  // bm1_y: broadcast learned scalar to [N,8] (output slot 3N..11N)
  {
    float bm = bm1g[0];
    float* bmout = out + 3 * (size_t)N + (size_t)blockIdx.x * (WAVES * 16 * 8);
    for (int i = tid; i < WAVES * 16 * 8; i += WAVES * 32) bmout[i] = bm;
  }

  if (r0 + 16 > N) return;

  __builtin_prefetch(Ig + (size_t)r0 * 8, 0, 3);
  __builtin_prefetch(Ug + (size_t)r0 * 8, 0, 3);

  v8f z13[4], z23[4], zs[4], z1s[4], z2s[4];

  // ================= pass 1: x = l1 (forward + reverse-mode grad) =================
  net_forward<true>(l1g, r0, ln, hi, s, ab, z13, z23, zs, z1s, z2s);
  finalDot(z13, z23, s + O_WF, ln, hi, out + N, r0);  // bl_1

  // VJP: v1 = (Wf .* z13) @ W2_3 + (Wf .* z23) @ W1_3
  storeScaleCol(ab, z13, s + O_WF, ln, hi);
  v8f v1[4];
  zero4(v1);
  gemm64(ab, s + O_W23T, ln, hi, v1);
  storeScaleCol(ab, z23, s + O_WF, ln, hi);
  gemm64(ab, s + O_W13T, ln, hi, v1);

  // g = (v1 .* z1) @ W2_2  (accumulated into one 16x16 tile; cols 0..7 valid)
  storeProd(ab, v1, z1s, ln, hi, 1.0f);
  v8f g;
#pragma unroll
  for (int j = 0; j < 8; ++j) g[j] = 0.0f;
  gemmN1(ab, s + O_W22T, ln, hi, g);

  // v2b = (v1 .* z2) @ W1_2 ; g += (2*v2b .* z) @ W1_1
  storeProd(ab, v1, z2s, ln, hi, 1.0f);
  v8f v2b[4];
  zero4(v2b);
  gemm64(ab, s + O_W12T, ln, hi, v2b);
  storeProd(ab, v2b, zs, ln, hi, 2.0f);
  gemmN1(ab, s + O_W11T, ln, hi, g);

  gradDot(g, ldotg, r0, ln, hi, out + 2 * (size_t)N);  // b1_grad

  // ================= pass 2: x = I =================
  net_forward<false>(Ig, r0, ln, hi, s, ab, z13, z23, nullptr, nullptr, nullptr);
  finalDot(z13, z23, s + O_WF, ln, hi, out, r0);  // b1_y

  // ================= pass 3: x = U =================
  net_forward<false>(Ug, r0, ln, hi, s, ab, z13, z23, nullptr, nullptr, nullptr);
  finalDot(z13, z23, s + O_WF, ln, hi, out + 11 * (size_t)N, r0);  // b2_y
}

}  // namespace

extern "C" void kernel_launch(void* const* d_in, const int* in_sizes, int n_in,
                              void* d_out, int out_size, void* d_ws, size_t ws_size,
                              hipStream_t stream) {
  const float* l1 = (const float*)d_in[0];
  const float* I = (const float*)d_in[1];
  const float* U = (const float*)d_in[2];
  const float* ldot = (const float*)d_in[3];
  const float* W11 = (const float*)d_in[4];
  const float* B11 = (const float*)d_in[5];
  const float* W12 = (const float*)d_in[6];
  const float* B12 = (const float*)d_in[7];
  const float* W22 = (const float*)d_in[8];
  const float* B22 = (const float*)d_in[9];
  const float* W13 = (const float*)d_in[10];
  const float* B13 = (const float*)d_in[11];
  const float* W23 = (const float*)d_in[12];
  const float* B23 = (const float*)d_in[13];
  const float* Wf = (const float*)d_in[14];
  const float* bm1 = (const float*)d_in[15];

  const int N = in_sizes[0] / 8;
  const int blocks = N / (WAVES * 16);
  const size_t shmem = (size_t)SMEM_FLOATS * sizeof(float);

  (void)hipFuncSetAttribute((const void*)pinn_fused,
                            hipFuncAttributeMaxDynamicSharedMemorySize,
                            (int)shmem);

  pinn_fused<<<blocks, WAVES * 32, shmem, stream>>>(
      l1, I, U, ldot, W11, B11, W12, B12, W22, B22, W13, B13, W23, B23, Wf,
      bm1, (float*)d_out, N);
}